// SASBlock_40939628265524
// MI455X (gfx1250) — compile-verified
//
#include <hip/hip_runtime.h>
#include <math.h>

typedef __attribute__((ext_vector_type(2))) float v2f;
typedef __attribute__((ext_vector_type(4))) float v4f;
typedef __attribute__((ext_vector_type(8))) float v8f;
typedef __attribute__((address_space(3))) float lds_f;

#define BB   4
#define NSEQ 2048
#define HID  1024
#define NH   16
#define DH   64
#define MROW (BB*NSEQ)      // 8192
#define NEGV -1.0e9f

__device__ __forceinline__ v8f wmma4(v2f a, v2f b, v8f c) {
    // D = A(16x4 f32) * B(4x16 f32) + C(16x16 f32)
    return __builtin_amdgcn_wmma_f32_16x16x4_f32(
        /*neg_a=*/false, a, /*neg_b=*/false, b,
        /*c_mod=*/(short)0, c, /*reuse_a=*/false, /*reuse_b=*/false);
}

// Order LDS ops within a wave (cross-lane exchange without workgroup barrier:
// DS ops from one wave complete in order; we just need the wait + no reordering).
__device__ __forceinline__ void wave_lds_sync() {
    __builtin_amdgcn_wave_barrier();
    asm volatile("s_wait_dscnt 0x0" ::: "memory");
    __builtin_amdgcn_wave_barrier();
}

__device__ __forceinline__ void wait_async0() {
    asm volatile("s_wait_asynccnt 0x0" ::: "memory");
}

// Direct global -> LDS async copy (16B per lane), tracked by ASYNCcnt.
__device__ __forceinline__ void async_g2l_b128(float* lds_dst,
                                               const float* sbase,
                                               unsigned vbyteoff) {
    lds_f* dst = (lds_f*)lds_dst;   // addrspacecast: 32-bit LDS offset
    asm volatile("global_load_async_to_lds_b128 %0, %1, %2"
                 :: "v"(dst), "v"(vbyteoff), "s"(sbase) : "memory");
}

// ---------------- LayerNorm (+ optional sign of sum|h|) ----------------
__global__ __launch_bounds__(128) void ln_kernel(
    const float* __restrict__ x, const float* __restrict__ g,
    const float* __restrict__ b, float* __restrict__ out,
    float* __restrict__ sgn)
{
    __shared__ float r1[128];
    __shared__ float r2[128];
    const int row = blockIdx.x;
    const int tid = threadIdx.x;
    const float* xr = x + (size_t)row * HID;

    float xv[8];
    float s = 0.f, s2 = 0.f;
#pragma unroll
    for (int i = 0; i < 8; ++i) {
        xv[i] = xr[tid + i * 128];
        s += xv[i];
        s2 += xv[i] * xv[i];
    }
    r1[tid] = s; r2[tid] = s2;
    __syncthreads();
    for (int off = 64; off > 0; off >>= 1) {
        if (tid < off) { r1[tid] += r1[tid + off]; r2[tid] += r2[tid + off]; }
        __syncthreads();
    }
    const float mean = r1[0] * (1.0f / HID);
    const float var  = r2[0] * (1.0f / HID) - mean * mean;
    const float rstd = rsqrtf(var + 1e-8f);
    __syncthreads();

    float asum = 0.f;
#pragma unroll
    for (int i = 0; i < 8; ++i) {
        const int c = tid + i * 128;
        const float hv = (xv[i] - mean) * rstd * g[c] + b[c];
        out[(size_t)row * HID + c] = hv;
        asum += fabsf(hv);
    }
    if (sgn) {
        r1[tid] = asum;
        __syncthreads();
        for (int off = 64; off > 0; off >>= 1) {
            if (tid < off) r1[tid] += r1[tid + off];
            __syncthreads();
        }
        if (tid == 0) sgn[row] = (r1[0] > 0.f) ? 1.f : 0.f;
    }
}

// ---------------- fp32 WMMA GEMM: C[M=8192, 1024] = A @ W (+epilogues) ------
// mode 0: out = A@W + bias, permuted to [B,NH,N,DH]
// mode 1: out = relu(A@W + bias)                    (row-major [M,HID])
// mode 2: out = x + eps*(A@W + bias + f)            (row-major [M,HID])
__global__ __launch_bounds__(256) void gemm_kernel(
    const float* __restrict__ A, const float* __restrict__ W,
    const float* __restrict__ bias, float* __restrict__ out,
    const float* __restrict__ xres, const float* __restrict__ fres,
    const float* __restrict__ epsp, int mode)
{
    __shared__ __align__(16) float As[32 * 36];
    __shared__ __align__(16) float Wt[64 * 34];   // transposed: Wt[col][k]

    const int tid  = threadIdx.x;
    const int m0   = blockIdx.x * 32;
    const int n0b  = blockIdx.y * 64;
    const int w    = tid >> 5;
    const int lane = tid & 31;
    const int l15  = lane & 15;
    const int hf   = lane >> 4;
    const int mw   = w >> 2;          // 0..1
    const int nw   = w & 3;           // 0..3

    // per-thread staging coordinates
    const int arow = tid >> 3, ac4 = (tid & 7) * 4;          // A: 32x32 chunk
    const unsigned avoff = (unsigned)((arow * HID + ac4) * sizeof(float));

    v8f acc = {};

    for (int k0 = 0; k0 < HID; k0 += 32) {
        // stage A chunk [32 x 32]: direct async global -> LDS (ASYNCcnt)
        async_g2l_b128(As + arow * 36 + ac4, A + (size_t)m0 * HID + k0, avoff);

        if (k0 + 32 < HID)   // prefetch next W panel line for this thread
            __builtin_prefetch(W + (size_t)(k0 + 32 + (tid >> 4)) * HID + n0b, 0, 0);

#pragma unroll
        for (int ss = 0; ss < 2; ++ss) {   // stage W chunk [32 x 64], transposed
            const int s  = tid + ss * 256;
            const int kr = s >> 4, c4 = (s & 15) * 4;
            v4f w4 = *(const v4f*)(W + (size_t)(k0 + kr) * HID + n0b + c4);
            Wt[(c4 + 0) * 34 + kr] = w4.x;
            Wt[(c4 + 1) * 34 + kr] = w4.y;
            Wt[(c4 + 2) * 34 + kr] = w4.z;
            Wt[(c4 + 3) * 34 + kr] = w4.w;
        }
        wait_async0();
        __syncthreads();
#pragma unroll
        for (int c = 0; c < 8; ++c) {
            v2f a = *(const v2f*)(As + (16 * mw + l15) * 36 + 4 * c + 2 * hf);
            v2f b = *(const v2f*)(Wt + (16 * nw + l15) * 34 + 4 * c + 2 * hf);
            acc = wmma4(a, b, acc);
        }
        __syncthreads();
    }

    const int cg = n0b + 16 * nw + l15;
    const float bv = bias[cg];
    const float eps0 = (mode == 2) ? epsp[0] : 0.f;
#pragma unroll
    for (int r = 0; r < 8; ++r) {
        const int rg = m0 + 16 * mw + r + 8 * hf;
        float val = acc[r] + bv;
        if (mode == 0) {
            const int bat = rg >> 11, n = rg & 2047;
            const int head = cg >> 6, d = cg & 63;
            out[(((size_t)(bat * NH + head)) * NSEQ + n) * DH + d] = val;
        } else if (mode == 1) {
            out[(size_t)rg * HID + cg] = fmaxf(val, 0.f);
        } else {
            const size_t idx = (size_t)rg * HID + cg;
            out[idx] = xres[idx] + eps0 * (val + fres[idx]);
        }
    }
}

// ---------------- Flash attention: one wave per (b*head, 16-query tile) ----
__global__ __launch_bounds__(128) void attn_kernel(
    const float* __restrict__ q, const float* __restrict__ k,
    const float* __restrict__ v, const float* __restrict__ sgn,
    const float* __restrict__ h, float* __restrict__ o)
{
    __shared__ __align__(16) float lds[4 * 320];   // per wave: st[16*18] + sc[16]
    const int wv   = threadIdx.x >> 5;
    const int lane = threadIdx.x & 31;
    const int l15  = lane & 15;
    const int hf   = lane >> 4;
    float* st = lds + wv * 320;       // S/P tile, stride 18
    float* sc = st + 288;             // per-row scale / combine factors

    const int wid   = blockIdx.x * 4 + wv;
    const int bh    = wid >> 7;             // 0..63
    const int itile = wid & 127;            // 0..127
    const int bat   = bh >> 4;
    const int head  = bh & 15;
    const int ibase = itile * 16;

    // Q tile (16x64) in registers, A-matrix layout per K=4 chunk
    const float* qrow = q + ((size_t)bh * NSEQ + ibase + l15) * DH;
    v2f qa[16];
#pragma unroll
    for (int c = 0; c < 16; ++c)
        qa[c] = *(const v2f*)(qrow + 4 * c + 2 * hf);

    v8f oacc[4] = {};
    float mrow = -3.0e38f, lrow = 0.f;

    const float* kbase = k + (size_t)bh * NSEQ * DH;
    const float* vbase = v + (size_t)bh * NSEQ * DH;
    const float* sgb   = sgn + bat * NSEQ;

    for (int j = 0; j <= itile; ++j) {
        const int jbase = j * 16;
        // S = Q @ K^T  (B operand = K^T: contiguous b64 from K rows)
        const float* krow = kbase + (size_t)(jbase + l15) * DH;
        v8f s = {};
#pragma unroll
        for (int c = 0; c < 16; ++c)
            s = wmma4(qa[c], *(const v2f*)(krow + 4 * c + 2 * hf), s);

        const float ks = sgb[jbase + l15];
        const bool diag = (j == itile);
#pragma unroll
        for (int r = 0; r < 8; ++r) {
            float val = s[r] * 0.125f;            // 1/sqrt(DH)
            const int m = r + 8 * hf;
            if (ks == 0.f) val = NEGV;            // key sign mask
            if (diag && l15 > m) val = NEGV;      // causal mask
            st[m * 18 + l15] = val;
        }
        wave_lds_sync();

        if (lane < 16) {                           // online softmax row stats
            float rm = st[lane * 18];
#pragma unroll
            for (int n = 1; n < 16; ++n) rm = fmaxf(rm, st[lane * 18 + n]);
            const float nm  = fmaxf(mrow, rm);
            const float scl = __expf(mrow - nm);
            float rs = 0.f;
#pragma unroll
            for (int n = 0; n < 16; ++n) {
                const float p = __expf(st[lane * 18 + n] - nm);
                st[lane * 18 + n] = p;
                rs += p;
            }
            lrow = lrow * scl + rs;
            mrow = nm;
            sc[lane] = scl;
        }
        wave_lds_sync();

        float sarr[8];
        *(v4f*)(sarr)     = *(const v4f*)(sc + 8 * hf);
        *(v4f*)(sarr + 4) = *(const v4f*)(sc + 8 * hf + 4);
#pragma unroll
        for (int g = 0; g < 4; ++g)
#pragma unroll
            for (int r = 0; r < 8; ++r) oacc[g][r] *= sarr[r];

        // O += P @ V
#pragma unroll
        for (int c = 0; c < 4; ++c) {
            v2f a = *(const v2f*)(st + l15 * 18 + 4 * c + 2 * hf);
            const float* vr = vbase + (size_t)(jbase + 4 * c + 2 * hf) * DH + l15;
#pragma unroll
            for (int g = 0; g < 4; ++g) {
                v2f b = { vr[16 * g], vr[16 * g + DH] };
                oacc[g] = wmma4(a, b, oacc[g]);
            }
        }
    }

    // epilogue: 1/L, query sign, residual (+h), write out
    if (lane < 16)
        sc[lane] = sgb[ibase + lane] / lrow;
    wave_lds_sync();
    float cf[8];
    *(v4f*)(cf)     = *(const v4f*)(sc + 8 * hf);
    *(v4f*)(cf + 4) = *(const v4f*)(sc + 8 * hf + 4);

#pragma unroll
    for (int g = 0; g < 4; ++g) {
#pragma unroll
        for (int r = 0; r < 8; ++r) {
            const int m = r + 8 * hf;
            const size_t idx = ((size_t)(bat * NSEQ + ibase + m)) * HID
                             + head * DH + 16 * g + l15;
            o[idx] = oacc[g][r] * cf[r] + h[idx];
        }
    }
}

extern "C" void kernel_launch(void* const* d_in, const int* in_sizes, int n_in,
                              void* d_out, int out_size, void* d_ws, size_t ws_size,
                              hipStream_t stream) {
    (void)in_sizes; (void)n_in; (void)out_size; (void)ws_size;
    const float* x    = (const float*)d_in[0];
    const float* wq   = (const float*)d_in[1];
    const float* wk   = (const float*)d_in[2];
    const float* wvp  = (const float*)d_in[3];
    const float* ffw1 = (const float*)d_in[4];
    const float* ffw2 = (const float*)d_in[5];
    const float* bq   = (const float*)d_in[6];
    const float* bk   = (const float*)d_in[7];
    const float* bvp  = (const float*)d_in[8];
    const float* ffb1 = (const float*)d_in[9];
    const float* ffb2 = (const float*)d_in[10];
    const float* ln1b = (const float*)d_in[11];
    const float* ln2b = (const float*)d_in[12];
    const float* ln1g = (const float*)d_in[13];
    const float* ln2g = (const float*)d_in[14];
    const float* eps  = (const float*)d_in[15];

    float* ws = (float*)d_ws;
    const size_t SZ = (size_t)MROW * HID;
    float* hbuf = ws;                 // LN1 output (later overlaid by attn out)
    float* qb   = ws + SZ;            // Q [B,NH,N,DH]   (later: FFN1 output)
    float* kb   = ws + 2 * SZ;        // K               (later: LN2 output)
    float* vb   = ws + 3 * SZ;        // V
    float* sgnb = ws + 4 * SZ;        // sign [B*N]
    float* obuf = hbuf;               // attn out overlays h (owner-wave RW)
    float* fbuf = kb;                 // LN2 out overlays K
    float* t1   = qb;                 // FFN1 out overlays Q

    const dim3 gg(MROW / 32, HID / 64), gb(256);

    ln_kernel<<<MROW, 128, 0, stream>>>(x, ln1g, ln1b, hbuf, sgnb);

    gemm_kernel<<<gg, gb, 0, stream>>>(hbuf, wq,  bq,  qb, nullptr, nullptr, nullptr, 0);
    gemm_kernel<<<gg, gb, 0, stream>>>(hbuf, wk,  bk,  kb, nullptr, nullptr, nullptr, 0);
    gemm_kernel<<<gg, gb, 0, stream>>>(hbuf, wvp, bvp, vb, nullptr, nullptr, nullptr, 0);

    attn_kernel<<<(BB * NH * (NSEQ / 16)) / 4, 128, 0, stream>>>(qb, kb, vb, sgnb, hbuf, obuf);

    ln_kernel<<<MROW, 128, 0, stream>>>(obuf, ln2g, ln2b, fbuf, nullptr);

    gemm_kernel<<<gg, gb, 0, stream>>>(fbuf, ffw1, ffb1, t1, nullptr, nullptr, nullptr, 1);
    gemm_kernel<<<gg, gb, 0, stream>>>(t1, ffw2, ffb2, (float*)d_out, x, fbuf, eps, 2);
}